// Regressor_7000796693085
// MI455X (gfx1250) — compile-verified
//
#include <hip/hip_runtime.h>
#include <hip/hip_bf16.h>

typedef __attribute__((ext_vector_type(2))) float v2f;
typedef __attribute__((ext_vector_type(8))) float v8f;

// D = A(16x4,f32) * B(4x16,f32) + C(16x16,f32)
#define WMMA_F32(a,b,c) __builtin_amdgcn_wmma_f32_16x16x4_f32(false,(a),false,(b),(short)0,(c),false,false)

static constexpr int Bn = 64, NJ = 21, Cc = 256, HW = 4096, H3 = 768;
static constexpr int MROWS = Bn * NJ; // 1344

// workspace layout (float offsets)
static constexpr size_t OFF_QPAD = 0;        // 64*32*256 = 524288
static constexpr size_t OFF_QB   = 524288;   // 1344*256  = 344064
static constexpr size_t OFF_XG   = 868352;   // 1344*768  = 1032192
static constexpr size_t OFF_H    = 1900544;  // 64*256    = 16384
static constexpr size_t OFF_YS   = 1916928;  // 344064
static constexpr size_t OFF_L    = 2260992;  // 441 (pad 512)
static constexpr size_t OFF_XL   = 2261504;  // 344064
static constexpr size_t OFF_T1   = 2605568;  // 344064
static constexpr size_t OFF_X2   = 2949632;  // 344064 -> total 3293696 floats (~12.6 MB)

__global__ void zero_kernel(float* __restrict__ p, int n) {
    int i = blockIdx.x * blockDim.x + threadIdx.x;
    if (i < n) p[i] = 0.0f;
}

// q[b,j,c] = sum_p target[b,j,p] * feature[b,c,p]
// grid = 64 blocks (one per batch), 256 threads = 8 waves; wave w owns N-strip [32w, 32w+32)
// each wave computes the full (padded) 32-row M with 4 WMMA accumulators -> feature read once.
__global__ void corr_kernel(const float* __restrict__ feat, const float* __restrict__ targ,
                            float* __restrict__ qpad) {
    const int b    = blockIdx.x;
    const int wave = threadIdx.x >> 5;
    const int lane = threadIdx.x & 31;
    const int r    = lane & 15;
    const int half = lane >> 4;
    const int n0   = wave * 32;

    const float* A  = targ + (size_t)b * NJ * HW;
    const float* Bm = feat + (size_t)b * Cc * HW;
    const float* ap0 = A  + (size_t)min(r,      NJ - 1) * HW;   // M-tile 0 (rows 0..15)
    const float* ap1 = A  + (size_t)min(16 + r, NJ - 1) * HW;   // M-tile 1 (rows 16..31, clamped)
    const float* bp0 = Bm + (size_t)(n0 + r)      * HW;
    const float* bp1 = Bm + (size_t)(n0 + 16 + r) * HW;

    v8f acc00 = {}, acc01 = {}, acc10 = {}, acc11 = {};
    const int base = 2 * half; // lanes 0-15 carry K {0,1}, lanes 16-31 carry K {2,3}

    #pragma unroll 8
    for (int k = 0; k < HW; k += 4) {
        const int off = k + base;
        v2f a0 = *(const v2f*)(ap0 + off);
        v2f a1 = *(const v2f*)(ap1 + off);
        v2f b0 = *(const v2f*)(bp0 + off);
        v2f b1 = *(const v2f*)(bp1 + off);
        acc00 = WMMA_F32(a0, b0, acc00);
        acc01 = WMMA_F32(a0, b1, acc01);
        acc10 = WMMA_F32(a1, b0, acc10);
        acc11 = WMMA_F32(a1, b1, acc11);
    }

    float* qb = qpad + (size_t)b * 32 * 256;
    #pragma unroll
    for (int v = 0; v < 8; v++) {
        const int row = v + 8 * half;
        qb[(size_t)row        * 256 + n0 + r]      = acc00[v];
        qb[(size_t)row        * 256 + n0 + 16 + r] = acc01[v];
        qb[(size_t)(16 + row) * 256 + n0 + r]      = acc10[v];
        qb[(size_t)(16 + row) * 256 + n0 + 16 + r] = acc11[v];
    }
}

// BatchNorm1d over joint dim (stats over batch & channel) + LeakyReLU(0.1)
__global__ void bn_kernel(const float* __restrict__ qpad, const float* __restrict__ gamma,
                          const float* __restrict__ beta, float* __restrict__ qb) {
    const int j = blockIdx.x;
    __shared__ float s1[256], s2[256];
    __shared__ float sc, sh;
    float s = 0.0f, ss = 0.0f;
    for (int i = threadIdx.x; i < Bn * Cc; i += 256) {
        const int b = i >> 8, c = i & 255;
        const float v = qpad[((size_t)b * 32 + j) * 256 + c];
        s += v; ss += v * v;
    }
    s1[threadIdx.x] = s; s2[threadIdx.x] = ss;
    __syncthreads();
    for (int st = 128; st > 0; st >>= 1) {
        if (threadIdx.x < st) { s1[threadIdx.x] += s1[threadIdx.x + st]; s2[threadIdx.x] += s2[threadIdx.x + st]; }
        __syncthreads();
    }
    if (threadIdx.x == 0) {
        const float inv_n = 1.0f / (float)(Bn * Cc);
        const float mean = s1[0] * inv_n;
        const float var  = s2[0] * inv_n - mean * mean;
        const float k = rsqrtf(var + 1e-5f) * gamma[j];
        sc = k; sh = beta[j] - mean * k;
    }
    __syncthreads();
    const float k = sc, d = sh;
    for (int i = threadIdx.x; i < Bn * Cc; i += 256) {
        const int b = i >> 8, c = i & 255;
        float v = qpad[((size_t)b * 32 + j) * 256 + c] * k + d;
        v = v > 0.0f ? v : 0.1f * v;
        qb[((size_t)b * NJ + j) * 256 + c] = v;
    }
}

// xg = qb[1344,256] @ wx[256,768] + bx   (WMMA)
__global__ void xg_kernel(const float* __restrict__ qb, const float* __restrict__ wx,
                          const float* __restrict__ bx, float* __restrict__ xg) {
    const int mt   = blockIdx.x;                 // 0..83
    const int wave = threadIdx.x >> 5;
    const int lane = threadIdx.x & 31;
    const int r = lane & 15, half = lane >> 4;
    const int n0  = (blockIdx.y * 8 + wave) * 16;
    const int col = n0 + r;
    const float* ap = qb + ((size_t)mt * 16 + r) * 256;
    const int base = 2 * half;
    v8f acc = {};
    #pragma unroll 8
    for (int k = 0; k < 256; k += 4) {
        const int off = k + base;
        v2f a = *(const v2f*)(ap + off);
        v2f bf = { wx[(size_t)off * H3 + col], wx[(size_t)(off + 1) * H3 + col] };
        acc = WMMA_F32(a, bf, acc);
    }
    const float bias = bx[col];
    #pragma unroll
    for (int v = 0; v < 8; v++) {
        const int row = mt * 16 + v + 8 * half;
        xg[(size_t)row * H3 + col] = acc[v] + bias;
    }
}

// One GRU step over the joint axis: hg = h@wh + bh (WMMA into LDS), then fused gates.
// grid = 4 blocks (16 batch rows each), 256 threads = 8 waves (6 N-tiles each).
__global__ void gru_step_kernel(const float* __restrict__ xg, const float* __restrict__ wh,
                                const float* __restrict__ bh, float* __restrict__ h,
                                float* __restrict__ ys, int t) {
    __shared__ float hg[16 * H3];  // 48 KB
    const int blk  = blockIdx.x;
    const int wave = threadIdx.x >> 5;
    const int lane = threadIdx.x & 31;
    const int r = lane & 15, half = lane >> 4;
    const float* ap = h + ((size_t)blk * 16 + r) * 256;
    const int base = 2 * half;

    for (int i = 0; i < 6; i++) {
        const int n0 = (wave * 6 + i) * 16;
        const int col = n0 + r;
        v8f acc = {};
        #pragma unroll 8
        for (int k = 0; k < 256; k += 4) {
            const int off = k + base;
            v2f a = *(const v2f*)(ap + off);
            v2f bf = { wh[(size_t)off * H3 + col], wh[(size_t)(off + 1) * H3 + col] };
            acc = WMMA_F32(a, bf, acc);
        }
        const float bias = bh[col];
        #pragma unroll
        for (int v = 0; v < 8; v++)
            hg[(v + 8 * half) * H3 + col] = acc[v] + bias;
    }
    __syncthreads();

    for (int i = threadIdx.x; i < 16 * 256; i += 256) {
        const int lb = i >> 8, c = i & 255;
        const int b = blk * 16 + lb;
        const float* xr = xg + ((size_t)b * NJ + t) * H3;
        const float* hr = hg + (size_t)lb * H3;
        const float rg = 1.0f / (1.0f + __expf(-(xr[c] + hr[c])));
        const float zg = 1.0f / (1.0f + __expf(-(xr[256 + c] + hr[256 + c])));
        const float nc = tanhf(xr[512 + c] + rg * hr[512 + c]);
        const float hold = h[(size_t)b * 256 + c];
        const float hn = (1.0f - zg) * nc + zg * hold;
        h[(size_t)b * 256 + c] = hn;
        ys[((size_t)b * NJ + t) * 256 + c] = hn;
    }
}

// L = I - D^{-1/2} A D^{-1/2}
__global__ void lap_kernel(const float* __restrict__ adj, float* __restrict__ L) {
    __shared__ float dinv[NJ];
    if (threadIdx.x < NJ) {
        float d = 0.0f;
        for (int m = 0; m < NJ; m++) d += adj[threadIdx.x * NJ + m];
        dinv[threadIdx.x] = d > 0.0f ? rsqrtf(d) : 0.0f;
    }
    __syncthreads();
    for (int i = threadIdx.x; i < NJ * NJ; i += blockDim.x) {
        const int n = i / NJ, m = i % NJ;
        float v = -dinv[n] * adj[i] * dinv[m];
        if (n == m) v += 1.0f;
        L[i] = v;
    }
}

// y[row,c] = sum_m L[n,m] * x[(b,m),c]    row = b*21 + n
__global__ void lx_kernel(const float* __restrict__ L, const float* __restrict__ x,
                          float* __restrict__ y) {
    const int row = blockIdx.x;
    const int c = threadIdx.x;
    const int n = row % NJ, b = row / NJ;
    const float* Lr = L + n * NJ;
    const float* xb = x + (size_t)b * NJ * 256 + c;
    float s = 0.0f;
    #pragma unroll
    for (int m = 0; m < NJ; m++) s += Lr[m] * xb[(size_t)m * 256];
    y[(size_t)row * 256 + c] = s;
}

// out = relu(x@w[0] + xl@w[1] + bias) [+ res]   (WMMA, two K=256 passes)
__global__ void cheb_kernel(const float* __restrict__ x, const float* __restrict__ xl,
                            const float* __restrict__ w, const float* __restrict__ bias,
                            const float* __restrict__ res, float* __restrict__ out) {
    const int mt   = blockIdx.x;
    const int wave = threadIdx.x >> 5;
    const int lane = threadIdx.x & 31;
    const int r = lane & 15, half = lane >> 4;
    const int n0  = (blockIdx.y * 8 + wave) * 16;
    const int col = n0 + r;
    const float* ap  = x  + ((size_t)mt * 16 + r) * 256;
    const float* ap2 = xl + ((size_t)mt * 16 + r) * 256;
    const float* w0 = w;
    const float* w1 = w + 256 * 256;
    const int base = 2 * half;
    v8f acc = {};
    #pragma unroll 8
    for (int k = 0; k < 256; k += 4) {
        const int off = k + base;
        v2f a = *(const v2f*)(ap + off);
        v2f bf = { w0[(size_t)off * 256 + col], w0[(size_t)(off + 1) * 256 + col] };
        acc = WMMA_F32(a, bf, acc);
    }
    #pragma unroll 8
    for (int k = 0; k < 256; k += 4) {
        const int off = k + base;
        v2f a = *(const v2f*)(ap2 + off);
        v2f bf = { w1[(size_t)off * 256 + col], w1[(size_t)(off + 1) * 256 + col] };
        acc = WMMA_F32(a, bf, acc);
    }
    const float bv = bias[col];
    #pragma unroll
    for (int v = 0; v < 8; v++) {
        const int row = mt * 16 + v + 8 * half;
        float val = acc[v] + bv;
        val = val > 0.0f ? val : 0.0f;
        if (res) val += res[(size_t)row * 256 + col];
        out[(size_t)row * 256 + col] = val;
    }
}

// final projection 256 -> 3 (tiny; scalar)
__global__ void out_kernel(const float* __restrict__ x, const float* __restrict__ xl,
                           const float* __restrict__ w, const float* __restrict__ bias,
                           float* __restrict__ out) {
    const int idx = blockIdx.x * blockDim.x + threadIdx.x;
    if (idx >= MROWS * 3) return;
    const int row = idx / 3, o = idx % 3;
    const float* w0 = w;            // [256,3]
    const float* w1 = w + 256 * 3;
    const float* xr  = x  + (size_t)row * 256;
    const float* xlr = xl + (size_t)row * 256;
    float s = bias[o];
    for (int c = 0; c < 256; c++)
        s += xr[c] * w0[c * 3 + o] + xlr[c] * w1[c * 3 + o];
    out[idx] = s;
}

extern "C" void kernel_launch(void* const* d_in, const int* in_sizes, int n_in,
                              void* d_out, int out_size, void* d_ws, size_t ws_size,
                              hipStream_t stream) {
    (void)in_sizes; (void)n_in; (void)out_size; (void)ws_size;
    const float* feat = (const float*)d_in[0];
    const float* targ = (const float*)d_in[1];
    const float* adj  = (const float*)d_in[2];
    const float* gam  = (const float*)d_in[3];
    const float* bet  = (const float*)d_in[4];
    const float* wx   = (const float*)d_in[5];
    const float* wh   = (const float*)d_in[6];
    const float* bx   = (const float*)d_in[7];
    const float* bhp  = (const float*)d_in[8];
    const float* wg1  = (const float*)d_in[9];
    const float* bg1  = (const float*)d_in[10];
    const float* wg2  = (const float*)d_in[11];
    const float* bg2  = (const float*)d_in[12];
    const float* wo   = (const float*)d_in[13];
    const float* bo   = (const float*)d_in[14];

    float* ws   = (float*)d_ws;
    float* qpad = ws + OFF_QPAD;
    float* qb   = ws + OFF_QB;
    float* xg   = ws + OFF_XG;
    float* h    = ws + OFF_H;
    float* ys   = ws + OFF_YS;
    float* L    = ws + OFF_L;
    float* xl   = ws + OFF_XL;
    float* t1   = ws + OFF_T1;
    float* x2   = ws + OFF_X2;

    zero_kernel<<<(16384 + 255) / 256, 256, 0, stream>>>(h, 16384);
    corr_kernel<<<64, 256, 0, stream>>>(feat, targ, qpad);
    bn_kernel<<<21, 256, 0, stream>>>(qpad, gam, bet, qb);
    xg_kernel<<<dim3(84, 6), 256, 0, stream>>>(qb, wx, bx, xg);
    for (int t = 0; t < NJ; t++)
        gru_step_kernel<<<4, 256, 0, stream>>>(xg, wh, bhp, h, ys, t);
    lap_kernel<<<1, 256, 0, stream>>>(adj, L);
    lx_kernel<<<MROWS, 256, 0, stream>>>(L, ys, xl);
    cheb_kernel<<<dim3(84, 2), 256, 0, stream>>>(ys, xl, wg1, bg1, nullptr, t1);
    lx_kernel<<<MROWS, 256, 0, stream>>>(L, t1, xl);
    cheb_kernel<<<dim3(84, 2), 256, 0, stream>>>(t1, xl, wg2, bg2, ys, x2);
    lx_kernel<<<MROWS, 256, 0, stream>>>(L, x2, xl);
    out_kernel<<<(MROWS * 3 + 255) / 256, 256, 0, stream>>>(x2, xl, wo, bo, (float*)d_out);
}